// Elastic_65429531787739
// MI455X (gfx1250) — compile-verified
//
#include <hip/hip_runtime.h>
#include <hip/hip_bf16.h>
#include <stdint.h>

// Problem constants (match reference)
constexpr int   NY    = 256;
constexpr int   NX    = 256;
constexpr int   NT    = 256;
constexpr int   B     = 2;
constexpr int   NSRC  = 2;
constexpr int   NREC  = 64;
constexpr float DY    = 5.0f;
constexpr float DX    = 5.0f;
constexpr float DT    = 0.0005f;
constexpr float PMLW  = 20.0f;

constexpr int NCELL = NY * NX;       // cells per batch
constexpr int NFTOT = B * NCELL;     // elements per field (batched)

// Workspace layout (floats):
//  [0 .. 13*NFTOT)     : vy vx syy sxy sxx m_syyy m_sxyx m_sxyy m_sxxx m_vyy m_vxx m_vyx m_vxy
//  [13*NFTOT .. +1024) : ay[256] by[256] ax[256] bx[256]
//  [+1024]             : max_vel (float bits, via atomicMax on uint)
constexpr size_t COEF_OFF = 13ull * NFTOT;
constexpr size_t MAXV_OFF = COEF_OFF + 4 * 256;

// ---------------- small 4-wide vector type (explicit, 16B aligned) ----------
struct __align__(16) V4 { float a0, a1, a2, a3; };

__device__ __forceinline__ V4 ld4(const float* p) { return *reinterpret_cast<const V4*>(p); }
__device__ __forceinline__ void st4(float* p, V4 v) { *reinterpret_cast<V4*>(p) = v; }
__device__ __forceinline__ V4 vadd(V4 a, V4 b) { return {a.a0+b.a0, a.a1+b.a1, a.a2+b.a2, a.a3+b.a3}; }
__device__ __forceinline__ V4 vsub(V4 a, V4 b) { return {a.a0-b.a0, a.a1-b.a1, a.a2-b.a2, a.a3-b.a3}; }
__device__ __forceinline__ V4 vmul(V4 a, V4 b) { return {a.a0*b.a0, a.a1*b.a1, a.a2*b.a2, a.a3*b.a3}; }
__device__ __forceinline__ V4 vmulf(V4 a, float s) { return {a.a0*s, a.a1*s, a.a2*s, a.a3*s}; }
// result[i] = c[i-1], with scalar feeding element 0 (left neighbor)
__device__ __forceinline__ V4 shl1(float left, V4 c) { return {left, c.a0, c.a1, c.a2}; }
// result[i] = c[i+1], with scalar feeding element 3 (right neighbor)
__device__ __forceinline__ V4 shr1(V4 c, float right) { return {c.a1, c.a2, c.a3, right}; }

// ---------------- CDNA5 async global->LDS staging helpers -------------------
__device__ __forceinline__ void async_g2lds_b128(unsigned lds_byte_addr, const float* gaddr) {
    // gfx1250: per-lane 16B async copy global -> LDS, tracked by ASYNCcnt
    asm volatile("global_load_async_to_lds_b128 %0, %1, off"
                 :: "v"(lds_byte_addr), "v"(gaddr) : "memory");
}
__device__ __forceinline__ void wait_async0() {
    asm volatile("s_wait_asynccnt 0" ::: "memory");
}
__device__ __forceinline__ unsigned lds_addr_of(const float* p) {
    // AS(3)->flat cast places the LDS offset in the low 32 bits
    return (unsigned)(uintptr_t)p;
}

// ---------------- max-velocity reduction ------------------------------------
__global__ __launch_bounds__(256) void maxvel_kernel(
    const float* __restrict__ lamb, const float* __restrict__ mu,
    const float* __restrict__ buoy, unsigned* __restrict__ maxvel_bits)
{
    __shared__ float red[256];
    int tid = threadIdx.x;
    int i = blockIdx.x * 256 + tid;
    float v = sqrtf((lamb[i] + 2.0f * mu[i]) * buoy[i]);
    red[tid] = v;
    __syncthreads();
    for (int s = 128; s > 0; s >>= 1) {
        if (tid < s) red[tid] = fmaxf(red[tid], red[tid + s]);
        __syncthreads();
    }
    if (tid == 0) atomicMax(maxvel_bits, __float_as_uint(red[0]));
}

// ---------------- PML coefficient profiles ----------------------------------
__global__ __launch_bounds__(256) void pml_kernel(
    const unsigned* __restrict__ maxvel_bits, float* __restrict__ coef)
{
    int i = threadIdx.x; // NY == NX == 256
    float max_vel = __uint_as_float(*maxvel_bits);
    float fi = (float)i;
    float f1 = (PMLW - fi) / PMLW;
    f1 = fminf(fmaxf(f1, 0.0f), 1.0f);
    float f2 = (fi - ((float)(NY - 1) - PMLW)) / PMLW;
    f2 = fminf(fmaxf(f2, 0.0f), 1.0f);
    float frac = fmaxf(f1, f2);
    // sigma_max = 3 * max_vel * ln(1000) / (2 * width * dx)
    float sigma_max = 3.0f * max_vel * 6.907755278982137f / (2.0f * PMLW * DX);
    float sigma = sigma_max * frac * frac;
    float alpha = 78.53981633974483f * (1.0f - frac); // pi * 25 * (1-frac)
    float bcoef = expf(-(sigma + alpha) * DT);
    float acoef = sigma / (sigma + alpha + 1e-9f) * (bcoef - 1.0f);
    coef[      i] = acoef;  // ay
    coef[256 + i] = bcoef;  // by
    coef[512 + i] = acoef;  // ax  (DY==DX, NY==NX -> identical profile)
    coef[768 + i] = bcoef;  // bx
}

// ---------------- phase 1: velocity + PML memories + sources ----------------
// block (64,4): each thread owns 4 consecutive x-cells (b128 traffic); one block = 4 full rows.
__global__ __launch_bounds__(256) void vel_step(
    float* __restrict__ ws, const float* __restrict__ buoy,
    const float* __restrict__ amps, const int* __restrict__ srcl, int t)
{
    __shared__ __align__(16) float s_ab[4 * 256]; // ay | by | ax | bx
    const float* coef = ws + COEF_OFF;
    int flat = threadIdx.y * 64 + threadIdx.x;    // 0..255
    async_g2lds_b128(lds_addr_of(&s_ab[flat * 4]), coef + flat * 4);
    wait_async0();
    __syncthreads();

    int tx = threadIdx.x;            // 0..63
    int x0 = tx * 4;
    int y  = blockIdx.y * 4 + threadIdx.y;
    int b  = blockIdx.z;
    int idx = b * NCELL + y * NX + x0;

    float*       vy     = ws;
    float*       vx     = ws + 1 * (size_t)NFTOT;
    const float* syy    = ws + 2 * (size_t)NFTOT;
    const float* sxy    = ws + 3 * (size_t)NFTOT;
    const float* sxx    = ws + 4 * (size_t)NFTOT;
    float*       m_syyy = ws + 5 * (size_t)NFTOT;
    float*       m_sxyx = ws + 6 * (size_t)NFTOT;
    float*       m_sxyy = ws + 7 * (size_t)NFTOT;
    float*       m_sxxx = ws + 8 * (size_t)NFTOT;

    const float inv_dy = 1.0f / DY, inv_dx = 1.0f / DX;
    float ayv = s_ab[y], byv = s_ab[256 + y];
    V4 ax4 = ld4(&s_ab[512 + x0]);
    V4 bx4 = ld4(&s_ab[768 + x0]);

    // d_syy_y = dby(syy) + PML memory
    V4 syy_c = ld4(syy + idx);
    V4 syy_u = (y > 0) ? ld4(syy + idx - NX) : syy_c;     // diff -> 0 at y==0
    V4 d = vmulf(vsub(syy_c, syy_u), inv_dy);
    V4 m = vadd(vmulf(ld4(m_syyy + idx), byv), vmulf(d, ayv));
    st4(m_syyy + idx, m);
    V4 d_syy_y = vadd(d, m);

    // d_sxy_x = dbx(sxy) + PML memory
    V4 sxy_c = ld4(sxy + idx);
    float sxy_l = (tx > 0) ? sxy[idx - 1] : sxy_c.a0;     // diff -> 0 at x==0
    d = vmulf(vsub(sxy_c, shl1(sxy_l, sxy_c)), inv_dx);
    m = vadd(vmul(ld4(m_sxyx + idx), bx4), vmul(d, ax4));
    st4(m_sxyx + idx, m);
    V4 d_sxy_x = vadd(d, m);

    V4 bv = ld4(buoy + y * NX + x0);
    V4 vy4 = vadd(ld4(vy + idx), vmul(vmulf(bv, DT), vadd(d_syy_y, d_sxy_x)));

    // d_sxy_y = dfy(sxy) + PML memory
    V4 sxy_d = (y < NY - 1) ? ld4(sxy + idx + NX) : sxy_c; // diff -> 0 at y==NY-1
    d = vmulf(vsub(sxy_d, sxy_c), inv_dy);
    m = vadd(vmulf(ld4(m_sxyy + idx), byv), vmulf(d, ayv));
    st4(m_sxyy + idx, m);
    V4 d_sxy_y = vadd(d, m);

    // d_sxx_x = dfx(sxx) + PML memory
    V4 sxx_c = ld4(sxx + idx);
    float sxx_r = (tx < 63) ? sxx[idx + 4] : sxx_c.a3;    // diff -> 0 at x==NX-1
    d = vmulf(vsub(shr1(sxx_c, sxx_r), sxx_c), inv_dx);
    m = vadd(vmul(ld4(m_sxxx + idx), bx4), vmul(d, ax4));
    st4(m_sxxx + idx, m);
    V4 d_sxx_x = vadd(d, m);

    V4 vx4 = vadd(ld4(vx + idx), vmul(vmulf(bv, DT), vadd(d_sxy_y, d_sxx_x)));
    st4(vx + idx, vx4);

    // point sources into vy (src_scale = DT * buoyancy[sy,sx])
    #pragma unroll
    for (int s = 0; s < NSRC; ++s) {
        int sy = srcl[(b * NSRC + s) * 2 + 0];
        int sx = srcl[(b * NSRC + s) * 2 + 1];
        if (y == sy && (sx >> 2) == tx) {
            float a = amps[(b * NSRC + s) * NT + t] * DT;
            int e = sx & 3;
            if      (e == 0) vy4.a0 += a * bv.a0;
            else if (e == 1) vy4.a1 += a * bv.a1;
            else if (e == 2) vy4.a2 += a * bv.a2;
            else             vy4.a3 += a * bv.a3;
        }
    }
    st4(vy + idx, vy4);
}

// ---------------- phase 2: stresses + PML memories + receivers --------------
__global__ __launch_bounds__(256) void str_step(
    float* __restrict__ ws, const float* __restrict__ lamb,
    const float* __restrict__ mu, const int* __restrict__ recl,
    float* __restrict__ out, int t)
{
    __shared__ __align__(16) float s_ab[4 * 256];
    const float* coef = ws + COEF_OFF;
    int flat = threadIdx.y * 64 + threadIdx.x;
    async_g2lds_b128(lds_addr_of(&s_ab[flat * 4]), coef + flat * 4);
    wait_async0();
    __syncthreads();

    int tx = threadIdx.x;
    int x0 = tx * 4;
    int y  = blockIdx.y * 4 + threadIdx.y;
    int b  = blockIdx.z;
    int idx = b * NCELL + y * NX + x0;

    const float* vy    = ws;
    const float* vx    = ws + 1 * (size_t)NFTOT;
    float*       syy   = ws + 2 * (size_t)NFTOT;
    float*       sxy   = ws + 3 * (size_t)NFTOT;
    float*       sxx   = ws + 4 * (size_t)NFTOT;
    float*       m_vyy = ws + 9  * (size_t)NFTOT;
    float*       m_vxx = ws + 10 * (size_t)NFTOT;
    float*       m_vyx = ws + 11 * (size_t)NFTOT;
    float*       m_vxy = ws + 12 * (size_t)NFTOT;

    const float inv_dy = 1.0f / DY, inv_dx = 1.0f / DX;
    float ayv = s_ab[y], byv = s_ab[256 + y];
    V4 ax4 = ld4(&s_ab[512 + x0]);
    V4 bx4 = ld4(&s_ab[768 + x0]);

    V4 vy_c = ld4(vy + idx);
    V4 vx_c = ld4(vx + idx);

    // d_vy_y = dfy(vy) + PML memory
    V4 vy_d = (y < NY - 1) ? ld4(vy + idx + NX) : vy_c;
    V4 d = vmulf(vsub(vy_d, vy_c), inv_dy);
    V4 m = vadd(vmulf(ld4(m_vyy + idx), byv), vmulf(d, ayv));
    st4(m_vyy + idx, m);
    V4 d_vy_y = vadd(d, m);

    // d_vx_x = dbx(vx) + PML memory
    float vx_l = (tx > 0) ? vx[idx - 1] : vx_c.a0;
    d = vmulf(vsub(vx_c, shl1(vx_l, vx_c)), inv_dx);
    m = vadd(vmul(ld4(m_vxx + idx), bx4), vmul(d, ax4));
    st4(m_vxx + idx, m);
    V4 d_vx_x = vadd(d, m);

    V4 la4 = ld4(lamb + y * NX + x0);
    V4 mu4 = ld4(mu + y * NX + x0);
    V4 l2m = vadd(la4, vmulf(mu4, 2.0f));

    V4 syy4 = vadd(ld4(syy + idx),
                   vmulf(vadd(vmul(l2m, d_vy_y), vmul(la4, d_vx_x)), DT));
    st4(syy + idx, syy4);
    V4 sxx4 = vadd(ld4(sxx + idx),
                   vmulf(vadd(vmul(la4, d_vy_y), vmul(l2m, d_vx_x)), DT));
    st4(sxx + idx, sxx4);

    // d_vy_x = dbx(vy) + PML memory
    float vy_l = (tx > 0) ? vy[idx - 1] : vy_c.a0;
    d = vmulf(vsub(vy_c, shl1(vy_l, vy_c)), inv_dx);
    m = vadd(vmul(ld4(m_vyx + idx), bx4), vmul(d, ax4));
    st4(m_vyx + idx, m);
    V4 d_vy_x = vadd(d, m);

    // d_vx_y = dby(vx) + PML memory
    V4 vx_u = (y > 0) ? ld4(vx + idx - NX) : vx_c;
    d = vmulf(vsub(vx_c, vx_u), inv_dy);
    m = vadd(vmulf(ld4(m_vxy + idx), byv), vmulf(d, ayv));
    st4(m_vxy + idx, m);
    V4 d_vx_y = vadd(d, m);

    V4 sxy4 = vadd(ld4(sxy + idx), vmul(vmulf(mu4, DT), vadd(d_vy_x, d_vx_y)));
    st4(sxy + idx, sxy4);

    // receivers: sample vy (unchanged by this phase) -> out[b, r, t]
    if (blockIdx.y == 0 && blockIdx.z == 0) {
        if (flat < B * NREC) {
            int rb = flat / NREC;
            int r  = flat % NREC;
            int ry = recl[(rb * NREC + r) * 2 + 0];
            int rx = recl[(rb * NREC + r) * 2 + 1];
            out[(rb * NREC + r) * NT + t] = vy[rb * NCELL + ry * NX + rx];
        }
    }
}

// ---------------- host-side launch ------------------------------------------
extern "C" void kernel_launch(void* const* d_in, const int* in_sizes, int n_in,
                              void* d_out, int out_size, void* d_ws, size_t ws_size,
                              hipStream_t stream) {
    const float* lamb = (const float*)d_in[0];
    const float* mu   = (const float*)d_in[1];
    const float* buoy = (const float*)d_in[2];
    const float* amps = (const float*)d_in[3];
    const int*   srcl = (const int*)d_in[4];
    const int*   recl = (const int*)d_in[5];
    float*       out  = (float*)d_out;
    float*       ws   = (float*)d_ws;

    // Zero fields + coefficient region + maxvel accumulator (deterministic per call).
    size_t total_bytes = (MAXV_OFF + 1) * sizeof(float);
    hipMemsetAsync(d_ws, 0, total_bytes, stream);

    unsigned* maxvel_bits = (unsigned*)(ws + MAXV_OFF);
    float*    coef        = ws + COEF_OFF;

    maxvel_kernel<<<(NY * NX) / 256, 256, 0, stream>>>(lamb, mu, buoy, maxvel_bits);
    pml_kernel<<<1, 256, 0, stream>>>(maxvel_bits, coef);

    dim3 blk(64, 4, 1);
    dim3 grd(1, NY / 4, B);   // 128 blocks per phase; thread = 4 cells (b128 traffic)
    for (int t = 0; t < NT; ++t) {
        vel_step<<<grd, blk, 0, stream>>>(ws, buoy, amps, srcl, t);
        str_step<<<grd, blk, 0, stream>>>(ws, lamb, mu, recl, out, t);
    }
}